// TestAllReduceFusedAddRMSNormStaticQuantFP4Model_7095285973071
// MI455X (gfx1250) — compile-verified
//
#include <hip/hip_runtime.h>
#include <stdint.h>

#define TOKENS   8192
#define HIDDEN   4096
#define NBLK     (HIDDEN / 16)        // 256 fp4-blocks per row
#define THREADS  256                  // 8 wave32s; one fp4-block per thread
#define GRID     1024
#define ROWS_PER_BLK (TOKENS / GRID)  // 8
#define EPS      1e-6f

typedef float f32x4 __attribute__((ext_vector_type(4)));

// Round positive float to FP8 E4M3 grid (saturate 448, min normal exp -6).
// floor(log2(mag)) computed exactly via frexp; rintf == jnp.round (half-even).
__device__ __forceinline__ float round_e4m3_pos(float mag) {
    int ep;
    (void)frexpf(fmaxf(mag, 1e-38f), &ep);   // mag = m * 2^ep, m in [0.5,1)
    float e = (float)(ep - 1);               // == floor(log2(mag)) exactly
    e = fminf(fmaxf(e, -6.0f), 8.0f);
    float step = ldexpf(1.0f, (int)e - 3);   // 3 mantissa bits
    float q = rintf(mag / step) * step;
    return fminf(q, 448.0f);
}

// Quantize to signed E2M1 grid {0,.5,1,1.5,2,3,4,6}; midpoints round up.
// Grid == uniform step 0.5 on [0,2), step 1 on [2,4), step 2 on [4,6].
// floor(m*rstep + 0.5) rounds-half-up; rstep is a power of two and m<=6,
// so the fma is exact at every tie point.
__device__ __forceinline__ float quant_e2m1(float x) {
    float m = fminf(fabsf(x), 6.0f);
    float rstep = (m < 2.0f) ? 2.0f : ((m < 4.0f) ? 1.0f : 0.5f);
    float step  = (m < 2.0f) ? 0.5f : ((m < 4.0f) ? 1.0f : 2.0f);
    float q = floorf(__builtin_fmaf(m, rstep, 0.5f)) * step;
    return copysignf(q, x);
}

__global__ __launch_bounds__(THREADS)
void fused_add_rmsnorm_nvfp4_kernel(const float* __restrict__ hs,
                                    const float* __restrict__ resid,
                                    const float* __restrict__ weight,
                                    const float* __restrict__ global_scale,
                                    float* __restrict__ out) {
    __shared__ float s_w[HIDDEN];                           // 16 KB staged weight
    __shared__ __align__(16) float s_red[2][THREADS / 32];  // double-buffered scratch

    const int tid = threadIdx.x;

    // ---- Stage weight row into LDS with CDNA5 async copies (ASYNCcnt) ----
    {
        const uint64_t wbase    = (uint64_t)(uintptr_t)weight;
        const uint32_t lds_base = (uint32_t)(uintptr_t)&s_w[0]; // low 32 bits of
                                                                // flat LDS addr == ds offset
        #pragma unroll
        for (int i = 0; i < HIDDEN / (4 * THREADS); ++i) {      // 4 iters x 16B/lane
            uint32_t byte_off = (uint32_t)(tid + i * THREADS) * 16u;
            uint32_t lds_addr = lds_base + byte_off;
            asm volatile("global_load_async_to_lds_b128 %0, %1, %2"
                         :: "v"(lds_addr), "v"(byte_off), "s"(wbase)
                         : "memory");
        }
        asm volatile("s_wait_asynccnt 0" ::: "memory");
    }
    __syncthreads();

    // Each thread uses the SAME 16 weights for every row it touches:
    // hoist them out of the row loop (one-time ds_load_b128 x4).
    f32x4 wv[4];
    #pragma unroll
    for (int i = 0; i < 4; ++i)
        wv[i] = *(const f32x4*)&s_w[tid * 16 + i * 4];

    const float gs = global_scale[0];

    float* __restrict__ q_out  = out;
    float* __restrict__ r_out  = out + (size_t)TOKENS * HIDDEN;
    float* __restrict__ sc_out = out + (size_t)2 * TOKENS * HIDDEN;

    for (int rr = 0; rr < ROWS_PER_BLK; ++rr) {
        const int row = blockIdx.x + rr * GRID;
        const size_t base = (size_t)row * HIDDEN + (size_t)tid * 16;

        // streamed add + resid_out store (non-temporal b128 both ways)
        f32x4 s[4];
        float ss = 0.0f;
        #pragma unroll
        for (int i = 0; i < 4; ++i) {
            f32x4 h = __builtin_nontemporal_load((const f32x4*)(hs    + base + i * 4));
            f32x4 r = __builtin_nontemporal_load((const f32x4*)(resid + base + i * 4));
            s[i] = h + r;
            ss += s[i].x * s[i].x + s[i].y * s[i].y
                + s[i].z * s[i].z + s[i].w * s[i].w;
            __builtin_nontemporal_store(s[i], (f32x4*)(r_out + base + i * 4));
        }

        // row sum-of-squares: wave32 xor-shuffle tree, then 8-wave LDS combine.
        // Double-buffered scratch => a single barrier per row is sufficient.
        #pragma unroll
        for (int off = 16; off > 0; off >>= 1)
            ss += __shfl_xor(ss, off, 32);
        const int par = rr & 1;
        if ((tid & 31) == 0) s_red[par][tid >> 5] = ss;
        __syncthreads();
        f32x4 red0 = *(const f32x4*)&s_red[par][0];    // ds_load_b128 x2
        f32x4 red1 = *(const f32x4*)&s_red[par][4];
        const float tot = (red0.x + red0.y) + (red0.z + red0.w)
                        + (red1.x + red1.y) + (red1.z + red1.w);

        const float rstd = rsqrtf(tot * (1.0f / HIDDEN) + EPS);

        // normalize with register-resident weight; per-16 amax (never crosses lanes)
        f32x4 n[4];
        float amax = 0.0f;
        #pragma unroll
        for (int i = 0; i < 4; ++i) {
            n[i] = s[i] * rstd * wv[i];
            amax = fmaxf(amax, fmaxf(fmaxf(fabsf(n[i].x), fabsf(n[i].y)),
                                     fmaxf(fabsf(n[i].z), fabsf(n[i].w))));
        }

        // NVFP4 block scale + quant
        const float sc  = round_e4m3_pos(amax * (1.0f / 6.0f) * gs);
        const float dq  = sc / gs;
        const float inv = (dq > 0.0f) ? (1.0f / dq) : 0.0f;    // inv==0 => q==0

        #pragma unroll
        for (int i = 0; i < 4; ++i) {
            f32x4 q;
            q.x = quant_e2m1(n[i].x * inv);
            q.y = quant_e2m1(n[i].y * inv);
            q.z = quant_e2m1(n[i].z * inv);
            q.w = quant_e2m1(n[i].w * inv);
            __builtin_nontemporal_store(q, (f32x4*)(q_out + base + i * 4));
        }
        __builtin_nontemporal_store(sc, sc_out + (size_t)row * NBLK + tid);
    }
}

extern "C" void kernel_launch(void* const* d_in, const int* in_sizes, int n_in,
                              void* d_out, int out_size, void* d_ws, size_t ws_size,
                              hipStream_t stream) {
    (void)in_sizes; (void)n_in; (void)out_size; (void)d_ws; (void)ws_size;
    const float* hs    = (const float*)d_in[0];
    const float* resid = (const float*)d_in[1];
    const float* w     = (const float*)d_in[2];
    const float* gs    = (const float*)d_in[3];
    float* out = (float*)d_out;
    fused_add_rmsnorm_nvfp4_kernel<<<GRID, THREADS, 0, stream>>>(hs, resid, w, gs, out);
}